// NoReductionTripletLoss_67010079752234
// MI455X (gfx1250) — compile-verified
//
#include <hip/hip_runtime.h>
#include <hip/hip_bf16.h>

typedef __attribute__((ext_vector_type(2))) float v2f;
typedef __attribute__((ext_vector_type(4))) float v4f;
typedef __attribute__((ext_vector_type(8))) float v8f;

#define N 256
#define K 2048
#define MARGIN 0.3f
#define EPS 1e-12f

// ---------------------------------------------------------------------------
// Kernel 1: per-row squared norms  sq[r] = sum_k X[r][k]^2
// ---------------------------------------------------------------------------
__global__ __launch_bounds__(256) void rowsq_kernel(const float* __restrict__ X,
                                                    float* __restrict__ sq) {
    __shared__ float red[256];
    const int r = blockIdx.x;
    const int t = threadIdx.x;
    const v4f* row = (const v4f*)(X + (size_t)r * K);   // 512 float4 per row
    float s = 0.0f;
    #pragma unroll
    for (int i = 0; i < 2; ++i) {
        v4f v = row[i * 256 + t];
        s += v.x * v.x + v.y * v.y + v.z * v.z + v.w * v.w;
    }
    red[t] = s;
    __syncthreads();
    for (int off = 128; off > 0; off >>= 1) {
        if (t < off) red[t] += red[t + off];
        __syncthreads();
    }
    if (t == 0) sq[r] = red[0];
}

// ---------------------------------------------------------------------------
// Kernel 2: Gram tile via V_WMMA_F32_16X16X4_F32, then distance matrix.
// One wave (32 threads) per 16x16 tile of G = X X^T.
//
// f32 16x4 A layout (ISA 7.12.2): lanes 0-15 = rows M, K-pos {0,1};
// lanes 16-31 = rows M, K-pos {2,3}.  B (4x16) mirrors with N in lanes.
// We remap K-position -> actual K so each lane reads a contiguous float4
// feeding two WMMAs:  iter 2t:   half h covers K = 8t+4h, 8t+4h+1
//                     iter 2t+1: half h covers K = 8t+4h+2, 8t+4h+3
// (a bijection over K; summation order is irrelevant to the result).
// ---------------------------------------------------------------------------
__global__ __launch_bounds__(32) void gram_dist_kernel(const float* __restrict__ X,
                                                       const float* __restrict__ sq,
                                                       float* __restrict__ dist) {
    const int tile = blockIdx.x;       // 0..255
    const int mt   = tile >> 4;
    const int nt   = tile & 15;
    const int lane = threadIdx.x;      // 0..31, all active (EXEC all-1s)
    const int half = lane >> 4;        // 0 or 1
    const int lm   = lane & 15;

    const float* arow = X + (size_t)(mt * 16 + lm) * K;
    const float* brow = X + (size_t)(nt * 16 + lm) * K;

    v8f c = {};
    for (int t = 0; t < K / 8; ++t) {
        const int base = t * 8 + half * 4;
        v4f a4 = *(const v4f*)(arow + base);
        v4f b4 = *(const v4f*)(brow + base);
        v2f a0 = { a4.x, a4.y };
        v2f b0 = { b4.x, b4.y };
        v2f a1 = { a4.z, a4.w };
        v2f b1 = { b4.z, b4.w };
        // 8 args: (neg_a, A, neg_b, B, c_mod, C, reuse_a, reuse_b)
        c = __builtin_amdgcn_wmma_f32_16x16x4_f32(false, a0, false, b0,
                                                  (short)0, c, false, false);
        c = __builtin_amdgcn_wmma_f32_16x16x4_f32(false, a1, false, b1,
                                                  (short)0, c, false, false);
    }

    // C/D layout: VGPR r, lanes 0-15 -> M = r, N = lane; lanes 16-31 -> M = 8+r.
    const int n   = nt * 16 + lm;
    const float sqn = sq[n];
    #pragma unroll
    for (int r = 0; r < 8; ++r) {
        const int m = mt * 16 + half * 8 + r;
        float d2 = sq[m] + sqn - 2.0f * c[r];
        d2 = fmaxf(d2, EPS);
        dist[m * N + n] = sqrtf(d2);
    }
}

// ---------------------------------------------------------------------------
// Kernel 3: stream the [N,N,N] output (the bandwidth-dominant pass).
// Block (a, chunk): caches dist row a + same-label mask in LDS, writes
// 8192 floats (2048 float4) of out[a, p, n].
// ---------------------------------------------------------------------------
__global__ __launch_bounds__(256) void triplet_kernel(const float* __restrict__ dist,
                                                      const long long* __restrict__ tgt,
                                                      float* __restrict__ out) {
    __shared__ float drow[N];
    __shared__ float samef[N];

    const int a     = blockIdx.x;
    const int chunk = blockIdx.y;       // 0..7
    const int t     = threadIdx.x;

    const long long la = tgt[a];
    drow[t]  = dist[a * N + t];
    samef[t] = (tgt[t] == la) ? 1.0f : 0.0f;
    __syncthreads();

    float* outa = out + (size_t)a * (N * N) + (size_t)chunk * 8192;

    #pragma unroll
    for (int i = 0; i < 8; ++i) {
        const int idx4 = i * 256 + t;            // float4 index within chunk
        const int flat = chunk * 8192 + idx4 * 4; // element index in [p][n]
        const int p  = flat >> 8;
        const int n0 = flat & 255;

        const float dap     = drow[p];
        const bool  same_ap = samef[p] != 0.0f;

        v4f v;
        #pragma unroll
        for (int j = 0; j < 4; ++j) {
            const int nn = n0 + j;
            const bool valid = same_ap && (samef[nn] == 0.0f);
            float tri = dap - drow[nn] + MARGIN;
            tri = fmaxf(tri, EPS);
            v[j] = valid ? tri : 0.0f;
        }
        *(v4f*)(outa + idx4 * 4) = v;            // global_store_b128
    }
}

// ---------------------------------------------------------------------------
extern "C" void kernel_launch(void* const* d_in, const int* in_sizes, int n_in,
                              void* d_out, int out_size, void* d_ws, size_t ws_size,
                              hipStream_t stream) {
    const float*     X   = (const float*)d_in[0];
    const long long* tgt = (const long long*)d_in[1];
    float*           out = (float*)d_out;

    float* sq   = (float*)d_ws;          // 256 floats
    float* dist = sq + N;                // 65536 floats

    rowsq_kernel<<<N, 256, 0, stream>>>(X, sq);
    gram_dist_kernel<<<N, 32, 0, stream>>>(X, sq, dist);
    dim3 grid3(N, 8);
    triplet_kernel<<<grid3, 256, 0, stream>>>(dist, tgt, out);
}